// sdfBoundaryModule_39402029973813
// MI455X (gfx1250) — compile-verified
//
#include <hip/hip_runtime.h>

// CDNA5 / gfx1250 (MI455X), wave32. Uses V_WMMA_F32_16X16X4_F32 for the
// affine geometry terms (ve, val3, |v|^2) of the SDF inner loop.

typedef __attribute__((ext_vector_type(2))) float v2f;
typedef __attribute__((ext_vector_type(8))) float v8f;

#define SUPPORT_F 0.05f
#define DH_F      1.0e-4f
#define KH_F      0.01f
#define EPS_F     1.0e-5f
#define L2_F      0.69314718055994530942f
#define INV_6PI_F 0.053051647697298445f

__device__ __forceinline__ float k_fn(float dr) {
  float ad = fabsf(dr);
  float d  = (ad < EPS_F) ? 1.0f : ad;
  float d2 = d * d, d3 = d * d2, d5 = d3 * d2, d7 = d5 * d2;
  float srt = sqrtf(fmaxf(1.0f - d2, 1.0e-12f));
  float dcl = fminf(fmaxf(d, 0.0f), 1.0f - 1.0e-7f);
  float c75 = 15.0f * d7 + 126.0f * d5;
  float integral = (-c75 * logf(2.0f * srt + 2.0f)
                    + c75 * logf(d)
                    + 6.0f * acosf(dcl)
                    + 15.0f * L2_F * d7
                    + srt * (97.0f * d5 + 60.0f * d3 - 22.0f * d)
                    + 126.0f * L2_F * d5) * INV_6PI_F;
  if (dr < 0.0f) integral = 1.0f - integral;
  if (ad < EPS_F) integral = 0.5f;
  float g = 1.0f - 0.5f * dr;
  return integral * (1.0f + g * g);
}

__global__ __launch_bounds__(256) void sdfBoundary_wmma_kernel(
    const float* __restrict__ poly,   // [16][2]
    const float* __restrict__ p,      // [M][2]
    float* __restrict__ out,          // c[M] | grad[M][2] | kint[M] | gk[M][2]
    int M, int tiles) {
  const int lane = threadIdx.x & 31;
  const int tile = blockIdx.x * 8 + (threadIdx.x >> 5);
  if (tile >= tiles) return;               // wave-uniform: EXEC all-ones below
  const bool hi = lane >= 16;
  const int  eA = lane & 15;

  // ---- A matrices (16 edges x K=4), ISA layout: lanes 0-15 hold K={0,1},
  //      lanes 16-31 hold K={2,3}, edge index M = lane&15.
  float ax = poly[2 * eA + 0], ay = poly[2 * eA + 1];
  int   e1 = (eA + 1) & 15;
  float bx = poly[2 * e1 + 0], by = poly[2 * e1 + 1];
  float ex = bx - ax, ey = by - ay;
  float cve = -(ax * ex + ay * ey);        // ve  = ex*px + ey*py + cve
  float c3  = ey * ax - ex * ay;           // v3  = -ey*px + ex*py + c3

  v2f A_ve, A_v3, A_vv;
  A_ve[0] = hi ? cve : ex;                 A_ve[1] = hi ? 0.0f : ey;
  A_v3[0] = hi ? c3  : -ey;                A_v3[1] = hi ? 0.0f : ex;
  A_vv[0] = hi ? (ax * ax + ay * ay) : (-2.0f * ax);   // vv = |p - a|^2
  A_vv[1] = hi ? 1.0f                : (-2.0f * ay);

  // ---- per-lane constants for the 8 D-matrix rows this lane owns:
  //      edge = r + 8*hi  (D layout: VGPR r -> M = r + 8*(lane>=16))
  float eeR[8], ieeR[8], pyR[8], pynR[8];
#pragma unroll
  for (int r = 0; r < 8; ++r) {
    int e  = r + (hi ? 8 : 0);
    int en = (e + 1) & 15;
    float pax = poly[2 * e + 0],  pay = poly[2 * e + 1];
    float pbx = poly[2 * en + 0], pby = poly[2 * en + 1];
    float exr = pbx - pax, eyr = pby - pay;
    float ee  = exr * exr + eyr * eyr;
    eeR[r]  = ee;
    ieeR[r] = 1.0f / ee;
    pyR[r]  = pay;                         // vy      = qy - poly[e].y
    pynR[r] = pby;                         // vy_next = qy - poly[e+1].y
  }

  // ---- B matrix (K=4 x 16 points): lanes 0-15 hold K={0,1}, 16-31 K={2,3},
  //      point column N = lane&15. Both halves load the same point.
  int pt  = tile * 16 + eA;
  int ptc = (pt < M) ? pt : (M - 1);
  float px = p[2 * ptc + 0];
  float py = p[2 * ptc + 1];

  const float offx[5] = {0.0f, DH_F, -DH_F, 0.0f, 0.0f};
  const float offy[5] = {0.0f, 0.0f, 0.0f, DH_F, -DH_F};
  float sd[5];

#pragma unroll
  for (int o = 0; o < 5; ++o) {
    float qx = px + offx[o];
    float qy = py + offy[o];
    v2f B;
    B[0] = hi ? 1.0f : qx;
    B[1] = hi ? (qx * qx + qy * qy) : qy;
    v8f z = {};
    v8f Dve = __builtin_amdgcn_wmma_f32_16x16x4_f32(false, A_ve, false, B,
                                                    (short)0, z, false, false);
    v8f Dv3 = __builtin_amdgcn_wmma_f32_16x16x4_f32(false, A_v3, false, B,
                                                    (short)0, z, false, false);
    v8f Dvv = __builtin_amdgcn_wmma_f32_16x16x4_f32(false, A_vv, false, B,
                                                    (short)0, z, false, false);

    float dmin = 3.402823466e38f;
    int wn = 0;
#pragma unroll
    for (int r = 0; r < 8; ++r) {
      float ve = Dve[r], v3 = Dv3[r], vv = Dvv[r];
      float t  = fminf(fmaxf(ve * ieeR[r], 0.0f), 1.0f);
      // |v - e t|^2 = vv - 2 t ve + t^2 ee
      float dist2 = vv - t * (2.0f * ve - t * eeR[r]);
      dmin = fminf(dmin, dist2);
      float vy  = qy - pyR[r];
      float vyn = qy - pynR[r];
      bool c1 = (vy  >= 0.0f);
      bool c2 = (vyn <  0.0f);
      wn += ( c1 &&  c2 && (v3 > 0.0f)) ? 1 : 0;
      wn -= (!c1 && !c2 && (v3 < 0.0f)) ? 1 : 0;
    }
    // merge the two lane-halves (each holds 8 of 16 edges for this point)
    dmin = fminf(dmin, __shfl_xor(dmin, 16, 32));
    wn  +=             __shfl_xor(wn,   16, 32);
    float s = (wn != 0) ? -1.0f : 1.0f;
    sd[o] = s * sqrtf(fmaxf(dmin, 0.0f));
  }

  // ---- finite-difference gradient + kernel integral
  float c  = sd[0];
  float gx = (sd[1] - sd[2]) * (0.5f / DH_F);
  float gy = (sd[3] - sd[4]) * (0.5f / DH_F);
  float gl = sqrtf(gx * gx + gy * gy);
  if (gl > EPS_F) { float inv = 1.0f / gl; gx *= inv; gy *= inv; }
  float cs   = c * (1.0f / SUPPORT_F);
  bool  mask = (cs <= 1.0f);
  float dr   = fminf(fmaxf(cs, 0.0f), 1.0f);
  float kc = k_fn(dr);
  float kr = k_fn(fminf(fmaxf(dr + KH_F, 0.0f), 1.0f));
  float kg = (kr - kc) * (1.0f / (KH_F * SUPPORT_F));
  float kint = mask ? kc : 0.0f;
  kg = mask ? kg : 0.0f;

  if (!hi && pt < M) {
    out[pt]                 = c;
    out[M + 2 * pt]         = gx;
    out[M + 2 * pt + 1]     = gy;
    out[3 * M + pt]         = kint;
    out[4 * M + 2 * pt]     = gx * kg;
    out[4 * M + 2 * pt + 1] = gy * kg;
  }
}

extern "C" void kernel_launch(void* const* d_in, const int* in_sizes, int n_in,
                              void* d_out, int out_size, void* d_ws, size_t ws_size,
                              hipStream_t stream) {
  (void)n_in; (void)out_size; (void)d_ws; (void)ws_size;
  const float* poly = (const float*)d_in[0];   // (16,2) f32
  const float* p    = (const float*)d_in[1];   // (M,2)  f32
  float* out = (float*)d_out;                  // 6*M f32
  int M = in_sizes[1] / 2;
  int tiles = (M + 15) / 16;                   // 16 points per wave
  int blocks = (tiles + 7) / 8;                // 8 waves (256 threads) per block
  sdfBoundary_wmma_kernel<<<blocks, 256, 0, stream>>>(poly, p, out, M, tiles);
}